// MaximumMeanDiscrepancy_1666447311298
// MI455X (gfx1250) — compile-verified
//
#include <hip/hip_runtime.h>

typedef __attribute__((ext_vector_type(2))) float v2f;
typedef __attribute__((ext_vector_type(4))) float v4f;
typedef __attribute__((ext_vector_type(8))) float v8f;

#define N_ROWS 8192
#define DIMS   16
#define INV_N2 (1.0f / 67108864.0f)   // 1 / (8192*8192)

// ---- pass 1: per-row squared norms for X (idx<8192) and Y (idx>=8192) ----
__global__ void mmd_norms(const float* __restrict__ X, const float* __restrict__ Y,
                          float* __restrict__ norms) {
    int idx = blockIdx.x * blockDim.x + threadIdx.x;      // 0..16383
    const float* src = (idx < N_ROWS) ? X : Y;
    int row = (idx < N_ROWS) ? idx : (idx - N_ROWS);
    const v4f* p = (const v4f*)(src + row * DIMS);
    v4f a = p[0], b = p[1], c = p[2], d = p[3];
    float s = 0.f;
#pragma unroll
    for (int k = 0; k < 4; ++k)
        s += a[k] * a[k] + b[k] * b[k] + c[k] * c[k] + d[k] * d[k];
    norms[idx] = s;
}

__global__ void mmd_zero(float* out) { out[0] = 0.f; }

// ---- pass 2: tiled RBF-mean accumulation via fp32 WMMA ----
// grid = (512 i-tiles, 8, 3 pairs), block = (32 lanes, 8 waves)
// pair: 0 = XX (+1), 1 = YY (+1), 2 = XY (-2)
__global__ void __launch_bounds__(256)
mmd_tiles(const float* __restrict__ X, const float* __restrict__ Y,
          const float* __restrict__ norms, float* __restrict__ out) {
    const int lane = threadIdx.x;          // 0..31
    const int half = lane >> 4;            // 0 or 1
    const int lm   = lane & 15;
    const int pair = blockIdx.z;

    const float* A  = (pair == 1) ? Y : X;
    const float* B  = (pair == 0) ? X : Y;
    const float* nA = norms + ((pair == 1) ? N_ROWS : 0);
    const float* nB = norms + ((pair == 0) ? 0 : N_ROWS);

    const int tile_i = blockIdx.x;                              // 0..511
    const int wj     = blockIdx.y * blockDim.y + threadIdx.y;   // 0..63

    // A operand: row (tile_i*16 + lm), features 4*kk + 2*half + {0,1}
    const float* pa = A + (tile_i * 16 + lm) * DIMS + 2 * half;
    v2f a0 = *(const v2f*)(pa + 0);
    v2f a1 = *(const v2f*)(pa + 4);
    v2f a2 = *(const v2f*)(pa + 8);
    v2f a3 = *(const v2f*)(pa + 12);

    // ||a||^2 for this lane's 8 output rows: M = r + 8*half
    v4f na_lo = *(const v4f*)(nA + tile_i * 16 + 8 * half);
    v4f na_hi = *(const v4f*)(nA + tile_i * 16 + 8 * half + 4);

    float s = 0.f;
#pragma unroll 1
    for (int t = 0; t < 8; ++t) {
        const int tile_j = wj * 8 + t;
        const float* pb = B + (tile_j * 16 + lm) * DIMS + 2 * half;
        v2f b0 = *(const v2f*)(pb + 0);
        v2f b1 = *(const v2f*)(pb + 4);
        v2f b2 = *(const v2f*)(pb + 8);
        v2f b3 = *(const v2f*)(pb + 12);
        float nb = nB[tile_j * 16 + lm];   // ||b||^2, N = lm

        v8f c = {};
        c = __builtin_amdgcn_wmma_f32_16x16x4_f32(false, a0, false, b0, (short)0, c, false, false);
        c = __builtin_amdgcn_wmma_f32_16x16x4_f32(false, a1, false, b1, (short)0, c, false, false);
        c = __builtin_amdgcn_wmma_f32_16x16x4_f32(false, a2, false, b2, (short)0, c, false, false);
        c = __builtin_amdgcn_wmma_f32_16x16x4_f32(false, a3, false, b3, (short)0, c, false, false);

#pragma unroll
        for (int r = 0; r < 8; ++r) {
            float na = (r < 4) ? na_lo[r] : na_hi[r - 4];
            float d2 = na + nb - 2.0f * c[r];
            d2 = fmaxf(d2, 0.0f);
            s += __expf(-d2);              // gamma = 1.0
        }
    }

    // wave32 reduction
#pragma unroll
    for (int off = 16; off > 0; off >>= 1)
        s += __shfl_down(s, off, 32);

    if (lane == 0) {
        float w = (pair == 2) ? -2.0f : 1.0f;
        atomicAdd(out, s * (w * INV_N2));
    }
}

extern "C" void kernel_launch(void* const* d_in, const int* in_sizes, int n_in,
                              void* d_out, int out_size, void* d_ws, size_t ws_size,
                              hipStream_t stream) {
    const float* X = (const float*)d_in[0];
    const float* Y = (const float*)d_in[1];
    float* out   = (float*)d_out;
    float* norms = (float*)d_ws;          // 16384 floats = 64 KB scratch

    mmd_zero<<<1, 1, 0, stream>>>(out);
    mmd_norms<<<(2 * N_ROWS) / 256, 256, 0, stream>>>(X, Y, norms);

    dim3 grid(N_ROWS / 16, 8, 3);         // 512 i-tiles, 8 j-groups, 3 pairs
    dim3 block(32, 8, 1);                 // 8 waves per block
    mmd_tiles<<<grid, block, 0, stream>>>(X, Y, norms, out);
}